// GAT_BCE_50981261804400
// MI455X (gfx1250) — compile-verified
//
#include <hip/hip_runtime.h>
#include <hip/hip_bf16.h>

#define HEADS 4
#define CPH 32
#define HC 128   // HEADS*CPH, also IN

typedef __attribute__((ext_vector_type(2))) float v2f;
typedef __attribute__((ext_vector_type(8))) float v8f;

// Only meaningful during the device pass; the host pass has no amdgcn builtins.
#if defined(__HIP_DEVICE_COMPILE__)
#if !__has_builtin(__builtin_amdgcn_wmma_f32_16x16x4_f32)
#error "wmma_f32_16x16x4_f32 builtin not available in device compilation"
#endif
#endif

// ---------------------------------------------------------------------------
// GEMM: Hout[N,128] = A[N,128] @ W[128,128] using V_WMMA_F32_16X16X4_F32.
// Block = 256 threads = 8 waves; block b computes row-tile b (16 rows),
// wave w computes column tile w (16 cols). W is staged in 64KB LDS.
// ---------------------------------------------------------------------------
__global__ __launch_bounds__(256) void gemm_wmma(
    const float* __restrict__ A, const float* __restrict__ W,
    float* __restrict__ Hout, int nrows) {
  __shared__ float sW[HC * HC];  // 64 KB
  for (int i = threadIdx.x; i < HC * HC; i += 256) sW[i] = W[i];
  __syncthreads();

  const int wave  = threadIdx.x >> 5;   // 0..7 -> column tile
  const int lane  = threadIdx.x & 31;
  const int trow  = blockIdx.x;         // row tile
  const int rbase = trow * 16;
  if (rbase >= nrows) return;           // wave-uniform

  const int m     = lane & 15;
  const int khalf = lane >> 4;          // 0: K pair {0,1}; 1: K pair {2,3}
  const int col   = wave * 16 + m;

  int row  = rbase + m;
  int lrow = row < nrows ? row : nrows - 1;  // clamp for ragged tail

  v8f acc = {};
#pragma unroll
  for (int k0 = 0; k0 < HC; k0 += 4) {
    const int ka = k0 + 2 * khalf;
    // A 16x4 layout: lane holds A[m][ka], A[m][ka+1] (8B aligned, ka even)
    v2f a = *(const v2f*)(A + (long long)lrow * HC + ka);
    // B 4x16 layout: lanes 0-15 hold rows K={k0,k0+1}, lanes 16-31 K={k0+2,k0+3}
    v2f b;
    b.x = sW[ka * HC + col];
    b.y = sW[(ka + 1) * HC + col];
    acc = __builtin_amdgcn_wmma_f32_16x16x4_f32(
        /*neg_a=*/false, a, /*neg_b=*/false, b,
        /*c_mod=*/(short)0, acc, /*reuse_a=*/false, /*reuse_b=*/false);
  }
  // D 16x16 layout: VGPR r -> M = r + 8*khalf, N = lane&15
#pragma unroll
  for (int r = 0; r < 8; ++r) {
    int orow = rbase + r + 8 * khalf;
    if (orow < nrows) Hout[(long long)orow * HC + col] = acc[r];
  }
}

// alpha_src[n,h] = sum_c h[n,h,c]*a_src[h,c]; same for a_dst (fused, h read once)
__global__ void node_alpha(const float* __restrict__ h,
                           const float* __restrict__ a_src,
                           const float* __restrict__ a_dst,
                           float* __restrict__ asrc, float* __restrict__ adst,
                           int n_nodes) {
  int idx = blockIdx.x * blockDim.x + threadIdx.x;  // n*4 + head
  if (idx >= n_nodes * HEADS) return;
  const int hd = idx & 3;
  const float* hp = h + (long long)(idx >> 2) * HC + hd * CPH;
  const float* as = a_src + hd * CPH;
  const float* ad = a_dst + hd * CPH;
  float s0 = 0.f, s1 = 0.f;
#pragma unroll
  for (int c = 0; c < CPH; ++c) {
    float v = hp[c];
    s0 += v * as[c];
    s1 += v * ad[c];
  }
  asrc[idx] = s0;
  adst[idx] = s1;
}

__global__ void fill_f32(float* __restrict__ p, float v, long long n) {
  long long i = (long long)blockIdx.x * blockDim.x + threadIdx.x;
  if (i < n) p[i] = v;
}

__device__ __forceinline__ void atomicMaxF(float* addr, float val) {
  if (val >= 0.f)
    atomicMax((int*)addr, __float_as_int(val));
  else
    atomicMin((unsigned int*)addr, (unsigned int)__float_as_int(val));
}

__device__ __forceinline__ void edge_endpoints(const int* __restrict__ ei,
                                               long long e, int E, int* s, int* d) {
  if (e < E) { *s = ei[e]; *d = ei[(long long)E + e]; }
  else       { *s = (int)(e - E); *d = *s; }   // self-loop
}

__device__ __forceinline__ float lrelu(float v) { return v > 0.f ? v : 0.2f * v; }

// Pass 1: segment max of leaky_relu logits over dst
__global__ void edge_max(const int* __restrict__ ei, int E, int n_nodes,
                         const float* __restrict__ asrc,
                         const float* __restrict__ adst,
                         float* __restrict__ m) {
  long long e = (long long)blockIdx.x * blockDim.x + threadIdx.x;
  if (e >= (long long)E + n_nodes) return;
  int s, d;
  edge_endpoints(ei, e, E, &s, &d);
#pragma unroll
  for (int hd = 0; hd < HEADS; ++hd)
    atomicMaxF(&m[d * HEADS + hd], lrelu(asrc[s * HEADS + hd] + adst[d * HEADS + hd]));
}

// Pass 2: denom[dst,h] += exp(e - m[dst,h])
__global__ void edge_expsum(const int* __restrict__ ei, int E, int n_nodes,
                            const float* __restrict__ asrc,
                            const float* __restrict__ adst,
                            const float* __restrict__ m,
                            float* __restrict__ denom) {
  long long e = (long long)blockIdx.x * blockDim.x + threadIdx.x;
  if (e >= (long long)E + n_nodes) return;
  int s, d;
  edge_endpoints(ei, e, E, &s, &d);
#pragma unroll
  for (int hd = 0; hd < HEADS; ++hd) {
    float v = lrelu(asrc[s * HEADS + hd] + adst[d * HEADS + hd]);
    atomicAdd(&denom[d * HEADS + hd], __expf(v - m[d * HEADS + hd]));
  }
}

// Pass 3: out[dst,:] += alpha * h[src,:]. 32 lanes per edge; lane covers
// channel (hd*32 + lane) for hd = 0..3. Gathers coalesced, scatter via f32 atomics.
__global__ void edge_msg(const int* __restrict__ ei, int E, int n_nodes,
                         const float* __restrict__ asrc,
                         const float* __restrict__ adst,
                         const float* __restrict__ m,
                         const float* __restrict__ denom,
                         const float* __restrict__ h,
                         float* __restrict__ out) {
  long long gid = (long long)blockIdx.x * blockDim.x + threadIdx.x;
  int lane = (int)(gid & 31);
  long long e = gid >> 5;
  if (e >= (long long)E + n_nodes) return;
  int s, d;
  edge_endpoints(ei, e, E, &s, &d);
  __builtin_prefetch(h + (long long)s * HC, 0, 3);  // global_prefetch source row
#pragma unroll
  for (int hd = 0; hd < HEADS; ++hd) {
    float v = lrelu(asrc[s * HEADS + hd] + adst[d * HEADS + hd]);
    float alpha = __expf(v - m[d * HEADS + hd]) / denom[d * HEADS + hd];
    int c = hd * CPH + lane;
    atomicAdd(&out[(long long)d * HC + c], h[(long long)s * HC + c] * alpha);
  }
}

// out += bias, then ELU, in place (result is next layer's input)
__global__ void node_epilogue(float* __restrict__ hA, const float* __restrict__ b,
                              long long n_elems) {
  long long i = (long long)blockIdx.x * blockDim.x + threadIdx.x;
  if (i >= n_elems) return;
  float v = hA[i] + b[(int)(i & (HC - 1))];
  hA[i] = v > 0.f ? v : __expf(v) - 1.f;
}

__global__ void pool_count(const int* __restrict__ batch, float* __restrict__ counts,
                           int n_nodes) {
  int n = blockIdx.x * blockDim.x + threadIdx.x;
  if (n < n_nodes) atomicAdd(&counts[batch[n]], 1.0f);
}

__global__ void pool_sum(const float* __restrict__ hA, const int* __restrict__ batch,
                         float* __restrict__ pooled, int n_nodes) {
  long long i = (long long)blockIdx.x * blockDim.x + threadIdx.x;
  if (i >= (long long)n_nodes * HC) return;
  int n = (int)(i >> 7), c = (int)(i & (HC - 1));
  atomicAdd(&pooled[batch[n] * HC + c], hA[i]);
}

__global__ void final_lin(const float* __restrict__ pooled,
                          const float* __restrict__ counts,
                          const float* __restrict__ lin_w,
                          const float* __restrict__ lin_b,
                          float* __restrict__ out, int G) {
  int g = blockIdx.x * blockDim.x + threadIdx.x;
  if (g >= G) return;
  float inv = 1.0f / fmaxf(counts[g], 1.0f);
  float acc = lin_b[0];
#pragma unroll 8
  for (int c = 0; c < HC; ++c) acc += pooled[g * HC + c] * inv * lin_w[c];
  out[g] = acc;
}

static inline unsigned blks(long long n, int t) { return (unsigned)((n + t - 1) / t); }

extern "C" void kernel_launch(void* const* d_in, const int* in_sizes, int n_in,
                              void* d_out, int out_size, void* d_ws, size_t ws_size,
                              hipStream_t stream) {
  const float* x     = (const float*)d_in[0];
  const int*   ei    = (const int*)d_in[1];
  const int*   batch = (const int*)d_in[2];
  const float* Wl[3]    = {(const float*)d_in[3],  (const float*)d_in[7],  (const float*)d_in[11]};
  const float* Asrc[3]  = {(const float*)d_in[4],  (const float*)d_in[8],  (const float*)d_in[12]};
  const float* Adst[3]  = {(const float*)d_in[5],  (const float*)d_in[9],  (const float*)d_in[13]};
  const float* Bias[3]  = {(const float*)d_in[6],  (const float*)d_in[10], (const float*)d_in[14]};
  const float* lin_w = (const float*)d_in[15];
  const float* lin_b = (const float*)d_in[16];
  float* out = (float*)d_out;

  const int N = in_sizes[0] / HC;
  const int E = in_sizes[1] / 2;
  const int G = out_size;
  const long long NE = (long long)E + N;   // edges incl. self-loops

  // workspace layout (floats)
  float* hA    = (float*)d_ws;                       // N*128: accumulator / layer input
  float* hB    = hA    + (size_t)N * HC;             // N*128: transformed features
  float* asrc  = hB    + (size_t)N * HC;             // N*4
  float* adst  = asrc  + (size_t)N * HEADS;          // N*4
  float* mbuf  = adst  + (size_t)N * HEADS;          // N*4
  float* denom = mbuf  + (size_t)N * HEADS;          // N*4
  float* pooled = denom + (size_t)N * HEADS;         // G*128
  float* counts = pooled + (size_t)G * HC;           // G

  const long long nfeat = (long long)N * HC;

  for (int l = 0; l < 3; ++l) {
    const float* in = (l == 0) ? x : hA;
    gemm_wmma<<<blks(N, 16), 256, 0, stream>>>(in, Wl[l], hB, N);
    node_alpha<<<blks((long long)N * HEADS, 256), 256, 0, stream>>>(
        hB, Asrc[l], Adst[l], asrc, adst, N);
    fill_f32<<<blks(nfeat, 256), 256, 0, stream>>>(hA, 0.0f, nfeat);
    fill_f32<<<blks((long long)N * HEADS, 256), 256, 0, stream>>>(
        mbuf, -INFINITY, (long long)N * HEADS);
    fill_f32<<<blks((long long)N * HEADS, 256), 256, 0, stream>>>(
        denom, 0.0f, (long long)N * HEADS);
    edge_max<<<blks(NE, 256), 256, 0, stream>>>(ei, E, N, asrc, adst, mbuf);
    edge_expsum<<<blks(NE, 256), 256, 0, stream>>>(ei, E, N, asrc, adst, mbuf, denom);
    edge_msg<<<blks(NE * 32, 256), 256, 0, stream>>>(
        ei, E, N, asrc, adst, mbuf, denom, hB, hA);
    node_epilogue<<<blks(nfeat, 256), 256, 0, stream>>>(hA, Bias[l], nfeat);
  }

  fill_f32<<<blks((long long)G * HC + G, 256), 256, 0, stream>>>(
      pooled, 0.0f, (long long)G * HC + G);  // pooled and counts are contiguous
  pool_count<<<blks(N, 256), 256, 0, stream>>>(batch, counts, N);
  pool_sum<<<blks(nfeat, 256), 256, 0, stream>>>(hA, batch, pooled, N);
  final_lin<<<1, ((G + 63) / 64) * 64, 0, stream>>>(pooled, counts, lin_w, lin_b, out, G);
}